// MultiHeadAttention_71528385347806
// MI455X (gfx1250) — compile-verified
//
#include <hip/hip_runtime.h>
#include <hip/hip_bf16.h>
#include <math.h>

#define NHEAD   8
#define DMODEL  256
#define DKDIM   32
#define DVDIM   32
#define LSEQ    1024
#define BATCH   2
#define LN_EPSF 1e-5f

typedef float v2f __attribute__((ext_vector_type(2)));
typedef float v8f __attribute__((ext_vector_type(8)));

// D = A(16x4,f32) * B(4x16,f32) + C, wave32.
// (neg_a, A, neg_b, B, c_mod, C, reuse_a, reuse_b)
__device__ __forceinline__ v8f wmma_f32_k4(v2f a, v2f b, v8f c) {
  return __builtin_amdgcn_wmma_f32_16x16x4_f32(false, a, false, b, (short)0, c,
                                               false, false);
}

// non-temporal loads — bias is streamed with zero reuse; keep it out of L2 so
// the 64MB attn matrix (and q/k/v/hn/emb) stay L2-resident.
__device__ __forceinline__ float ntload(const float* p) {
  return __builtin_nontemporal_load(p);
}
__device__ __forceinline__ v2f ntload2(const float* p) {
  return __builtin_nontemporal_load((const v2f*)p);  // 8B-aligned NT b64
}

// ---------------------------------------------------------------- LayerNorm
__global__ void ln_kernel(const float* __restrict__ h,
                          const float* __restrict__ gamma,
                          const float* __restrict__ beta,
                          float* __restrict__ hn) {
  int row  = blockIdx.x;
  int lane = threadIdx.x;
  const float* x = h + (size_t)row * DMODEL;
  float vals[8];
  float s = 0.f;
#pragma unroll
  for (int t = 0; t < 8; ++t) { vals[t] = x[lane + 32 * t]; s += vals[t]; }
#pragma unroll
  for (int o = 16; o > 0; o >>= 1) s += __shfl_xor(s, o, 32);
  float mu = s * (1.0f / DMODEL);
  float vs = 0.f;
#pragma unroll
  for (int t = 0; t < 8; ++t) { float d = vals[t] - mu; vs += d * d; }
#pragma unroll
  for (int o = 16; o > 0; o >>= 1) vs += __shfl_xor(vs, o, 32);
  float inv = rsqrtf(vs * (1.0f / DMODEL) + LN_EPSF);
  float* y = hn + (size_t)row * DMODEL;
#pragma unroll
  for (int t = 0; t < 8; ++t) {
    int c = lane + 32 * t;
    y[c] = (vals[t] - mu) * inv * gamma[c] + beta[c];
  }
}

// ------------------------------------------------- GEMM: C = A @ W^T (WMMA)
__global__ void proj_kernel(const float* __restrict__ A,
                            const float* __restrict__ W,
                            float* __restrict__ C) {
  int m0 = blockIdx.x * 16, n0 = blockIdx.y * 16;
  int lane = threadIdx.x, half = lane >> 4, l = lane & 15;
  const float* arow = A + (size_t)(m0 + l) * DMODEL;
  const float* wrow = W + (size_t)(n0 + l) * DMODEL;  // B[k][n] = W[n][k]
  v8f c = {};
  for (int k0 = 0; k0 < DMODEL; k0 += 4) {
    v2f a, b;
    a.x = arow[k0 + 2 * half]; a.y = arow[k0 + 2 * half + 1];
    b.x = wrow[k0 + 2 * half]; b.y = wrow[k0 + 2 * half + 1];
    c = wmma_f32_k4(a, b, c);
  }
#pragma unroll
  for (int r = 0; r < 8; ++r)
    C[(size_t)(m0 + r + 8 * half) * DMODEL + n0 + l] = c[r];
}

// --------------------------------------- attn1[h,b,i,j] = sum_d q_i.d k_j.d
__global__ void attn1_kernel(const float* __restrict__ Q,
                             const float* __restrict__ K,
                             float* __restrict__ attn) {
  int bh = blockIdx.x; int b = bh >> 3; int hh = bh & 7;
  int i0 = blockIdx.y * 16, j0 = blockIdx.z * 16;
  int lane = threadIdx.x, half = lane >> 4, l = lane & 15;
  const float* qrow = Q + ((size_t)(b * LSEQ + i0 + l) * NHEAD + hh) * DKDIM;
  const float* krow = K + ((size_t)(b * LSEQ + j0 + l) * NHEAD + hh) * DKDIM;
  v8f c = {};
#pragma unroll
  for (int k0 = 0; k0 < DKDIM; k0 += 4) {
    v2f a, bf;
    a.x  = qrow[k0 + 2 * half]; a.y  = qrow[k0 + 2 * half + 1];
    bf.x = krow[k0 + 2 * half]; bf.y = krow[k0 + 2 * half + 1];
    c = wmma_f32_k4(a, bf, c);
  }
  size_t base = ((size_t)hh * BATCH + b) * LSEQ * LSEQ;
#pragma unroll
  for (int r = 0; r < 8; ++r)
    attn[base + (size_t)(i0 + r + 8 * half) * LSEQ + j0 + l] = c[r];
}

// ---- attn2[h,b,i,j] += sum_d bias[b,i,j,d] q[b,i,h,d]   (per-(b,i) GEMM:
//      M = 16 j's, N = 8 heads (padded), K = 32; bias read ONCE, NT b64)
__global__ void attn2_kernel(const float* __restrict__ bias,
                             const float* __restrict__ Q,
                             float* __restrict__ attn) {
  int b = blockIdx.x, i = blockIdx.y, j0 = blockIdx.z * 16;
  int lane = threadIdx.x, half = lane >> 4, l = lane & 15;
  bool nvalid = l < 8;
  const float* arow = bias + (((size_t)b * LSEQ + i) * LSEQ + j0 + l) * DKDIM;
  const float* qv   = Q + ((size_t)(b * LSEQ + i) * NHEAD + (l & 7)) * DKDIM;
  v8f c = {};
#pragma unroll
  for (int k0 = 0; k0 < DKDIM; k0 += 4) {
    v2f a = ntload2(arow + k0 + 2 * half);
    v2f bf;
    if (nvalid) { bf.x = qv[k0 + 2 * half]; bf.y = qv[k0 + 2 * half + 1]; }
    else        { bf.x = 0.f;               bf.y = 0.f; }
    c = wmma_f32_k4(a, bf, c);
  }
  if (nvalid) {  // lane owns head l, 8 contiguous j's -> vector RMW
    float* pr = attn + (((size_t)l * BATCH + b) * LSEQ + i) * LSEQ + j0 + 8 * half;
    float4 x0 = *(float4*)(pr);
    float4 x1 = *(float4*)(pr + 4);
    x0.x += c[0]; x0.y += c[1]; x0.z += c[2]; x0.w += c[3];
    x1.x += c[4]; x1.y += c[5]; x1.z += c[6]; x1.w += c[7];
    *(float4*)(pr)     = x0;
    *(float4*)(pr + 4) = x1;
  }
}

// ----------------------------------------- softmax over j (scaled), in-place
__global__ void softmax_kernel(float* __restrict__ attn) {
  __shared__ float red[8];
  float4* p = (float4*)(attn + (size_t)blockIdx.x * LSEQ);
  int t = threadIdx.x;
  const float scale = 0.17677669529663687f;  // 1/sqrt(32)
  float4 vv = p[t];
  float v[4] = {vv.x * scale, vv.y * scale, vv.z * scale, vv.w * scale};
  float mx = fmaxf(fmaxf(v[0], v[1]), fmaxf(v[2], v[3]));
#pragma unroll
  for (int o = 16; o > 0; o >>= 1) mx = fmaxf(mx, __shfl_xor(mx, o, 32));
  int wid = t >> 5;
  if ((t & 31) == 0) red[wid] = mx;
  __syncthreads();
  float bm = red[0];
#pragma unroll
  for (int w = 1; w < 8; ++w) bm = fmaxf(bm, red[w]);
  float s = 0.f;
#pragma unroll
  for (int u = 0; u < 4; ++u) { v[u] = __expf(v[u] - bm); s += v[u]; }
#pragma unroll
  for (int o = 16; o > 0; o >>= 1) s += __shfl_xor(s, o, 32);
  __syncthreads();
  if ((t & 31) == 0) red[wid] = s;
  __syncthreads();
  float tot = 0.f;
#pragma unroll
  for (int w = 0; w < 8; ++w) tot += red[w];
  float inv = 1.0f / tot;
  p[t] = make_float4(v[0] * inv, v[1] * inv, v[2] * inv, v[3] * inv);
}

// ---- emb1[b,i,h,d] = sum_j attn[h,b,i,j] v[b,j,h,d]; both d-halves per block
__global__ void emb1_kernel(const float* __restrict__ attn,
                            const float* __restrict__ V,
                            float* __restrict__ emb) {
  int bh = blockIdx.x; int b = bh >> 3; int hh = bh & 7;
  int i0 = blockIdx.y * 16;
  int lane = threadIdx.x, half = lane >> 4, l = lane & 15;
  const float* arow = attn + (((size_t)hh * BATCH + b) * LSEQ + i0 + l) * LSEQ;
  v8f c0 = {}, c1 = {};
  for (int k0 = 0; k0 < LSEQ; k0 += 4) {
    v2f a;
    a.x = arow[k0 + 2 * half]; a.y = arow[k0 + 2 * half + 1];
    const float* v0 =
        V + ((size_t)(b * LSEQ + k0 + 2 * half) * NHEAD + hh) * DVDIM + l;
    v2f b0, b1;
    b0.x = v0[0];  b0.y = v0[NHEAD * DVDIM];
    b1.x = v0[16]; b1.y = v0[NHEAD * DVDIM + 16];
    c0 = wmma_f32_k4(a, b0, c0);
    c1 = wmma_f32_k4(a, b1, c1);
  }
  float* e = emb + (size_t)(b * LSEQ + i0) * (NHEAD * DVDIM) + hh * DVDIM;
#pragma unroll
  for (int r = 0; r < 8; ++r) {
    e[(size_t)(r + 8 * half) * (NHEAD * DVDIM) + l]      = c0[r];
    e[(size_t)(r + 8 * half) * (NHEAD * DVDIM) + 16 + l] = c1[r];
  }
}

// -- emb2[b,i,h,d] += sum_j attn[h,b,i,j] bias[b,i,j,d]  (bias NT-streamed)
__global__ void emb2_kernel(const float* __restrict__ attn,
                            const float* __restrict__ bias,
                            float* __restrict__ emb) {
  int bi = blockIdx.x; int b = bi >> 10; int i = bi & 1023;
  int lane = threadIdx.x, half = lane >> 4, l = lane & 15;
  bool mvalid = l < 8;  // A rows = heads
  const float* arow = attn + (((size_t)(l & 7) * BATCH + b) * LSEQ + i) * LSEQ;
  const float* brow = bias + ((size_t)b * LSEQ + i) * (size_t)LSEQ * DKDIM;
  v8f c0 = {}, c1 = {};
  for (int k0 = 0; k0 < LSEQ; k0 += 4) {
    v2f a;
    if (mvalid) { a.x = arow[k0 + 2 * half]; a.y = arow[k0 + 2 * half + 1]; }
    else        { a.x = 0.f;                 a.y = 0.f; }
    const float* p0 = brow + (size_t)(k0 + 2 * half) * DKDIM;
    v2f b0, b1;
    b0.x = ntload(p0 + l);              b0.y = ntload(p0 + DKDIM + l);
    b1.x = ntload(p0 + 16 + l);         b1.y = ntload(p0 + DKDIM + 16 + l);
    c0 = wmma_f32_k4(a, b0, c0);
    c1 = wmma_f32_k4(a, b1, c1);
  }
  if (half == 0) {  // output rows m = r — only heads 0..7 valid
    float* e = emb + (size_t)(b * LSEQ + i) * (NHEAD * DVDIM);
#pragma unroll
    for (int r = 0; r < 8; ++r) {
      e[r * DVDIM + l]      += c0[r];
      e[r * DVDIM + 16 + l] += c1[r];
    }
  }
}

// -------------------- out = emb @ fc_w^T + fc_b + residual (WMMA + epilogue)
__global__ void fc_kernel(const float* __restrict__ emb,
                          const float* __restrict__ W,
                          const float* __restrict__ bvec,
                          const float* __restrict__ resid,
                          float* __restrict__ out) {
  int m0 = blockIdx.x * 16, n0 = blockIdx.y * 16;
  int lane = threadIdx.x, half = lane >> 4, l = lane & 15;
  const float* arow = emb + (size_t)(m0 + l) * DMODEL;
  const float* wrow = W + (size_t)(n0 + l) * DMODEL;
  v8f c = {};
  for (int k0 = 0; k0 < DMODEL; k0 += 4) {
    v2f a, bf;
    a.x  = arow[k0 + 2 * half]; a.y  = arow[k0 + 2 * half + 1];
    bf.x = wrow[k0 + 2 * half]; bf.y = wrow[k0 + 2 * half + 1];
    c = wmma_f32_k4(a, bf, c);
  }
#pragma unroll
  for (int r = 0; r < 8; ++r) {
    int m = m0 + r + 8 * half, n = n0 + l;
    out[(size_t)m * DMODEL + n] = c[r] + bvec[n] + resid[(size_t)m * DMODEL + n];
  }
}

extern "C" void kernel_launch(void* const* d_in, const int* in_sizes, int n_in,
                              void* d_out, int out_size, void* d_ws,
                              size_t ws_size, hipStream_t stream) {
  (void)in_sizes; (void)n_in; (void)out_size; (void)ws_size;
  const float* h    = (const float*)d_in[0];
  const float* bias = (const float*)d_in[1];
  const float* w_qs = (const float*)d_in[2];
  const float* w_ks = (const float*)d_in[3];
  const float* w_vs = (const float*)d_in[4];
  const float* ln_g = (const float*)d_in[5];
  const float* ln_b = (const float*)d_in[6];
  const float* fc_w = (const float*)d_in[7];
  const float* fc_b = (const float*)d_in[8];

  float* out  = (float*)d_out;                               // (B,L,256)
  float* attn = out + (size_t)BATCH * LSEQ * DMODEL;         // (H,B,L,L)

  const size_t ROWS = (size_t)BATCH * LSEQ;                  // 2048
  float* ws  = (float*)d_ws;                                 // 10 MB used
  float* hn  = ws;
  float* q   = hn + ROWS * DMODEL;
  float* k   = q + ROWS * DMODEL;
  float* v   = k + ROWS * DMODEL;
  float* emb = v + ROWS * DMODEL;

  ln_kernel<<<(int)ROWS, 32, 0, stream>>>(h, ln_g, ln_b, hn);

  dim3 gproj((unsigned)(ROWS / 16), DMODEL / 16);
  proj_kernel<<<gproj, 32, 0, stream>>>(hn, w_qs, q);
  proj_kernel<<<gproj, 32, 0, stream>>>(hn, w_ks, k);
  proj_kernel<<<gproj, 32, 0, stream>>>(hn, w_vs, v);

  attn1_kernel<<<dim3(BATCH * NHEAD, LSEQ / 16, LSEQ / 16), 32, 0, stream>>>(q, k, attn);
  attn2_kernel<<<dim3(BATCH, LSEQ, LSEQ / 16), 32, 0, stream>>>(bias, q, attn);
  softmax_kernel<<<NHEAD * BATCH * LSEQ, 256, 0, stream>>>(attn);
  emb1_kernel<<<dim3(BATCH * NHEAD, LSEQ / 16), 32, 0, stream>>>(attn, v, emb);
  emb2_kernel<<<(unsigned)ROWS, 32, 0, stream>>>(attn, bias, emb);
  fc_kernel<<<gproj, 32, 0, stream>>>(emb, fc_w, fc_b, h, out);
}